// DiscriminatorImg_67035849556129
// MI455X (gfx1250) — compile-verified
//
#include <hip/hip_runtime.h>

typedef __attribute__((ext_vector_type(2))) float        v2f;
typedef __attribute__((ext_vector_type(8))) float        v8f;
typedef __attribute__((ext_vector_type(4))) unsigned int v4u;
typedef __attribute__((ext_vector_type(8))) int          v8i;
typedef __attribute__((ext_vector_type(4))) int          v4i;

#define IN_EPS 1e-5f

__device__ __forceinline__ float lrelu(float v) { return v > 0.f ? v : 0.01f * v; }

// -------- Fused conv2x2(stride2) + InstanceNorm + LeakyReLU -----------------
template<int CIN, int CTOT, int COUT, int HOUT, bool HAS_LAB>
__global__ __launch_bounds__(256) void conv_in_lrelu_kernel(
    const float* __restrict__ in,      // [B, CIN, 2*HOUT, 2*HOUT]
    const float* __restrict__ w,       // [COUT, CTOT, 2, 2]
    const float* __restrict__ bias,    // [COUT]
    const float* __restrict__ labels,  // [B, 20] or nullptr
    float* __restrict__ out)           // [B, COUT, HOUT, HOUT]
{
  constexpr int T   = 256;
  constexpr int HIN = 2 * HOUT;
  constexpr int P   = HOUT * HOUT;
  constexpr int VPT = (P + T - 1) / T;

  const int b   = blockIdx.x;
  const int tid = threadIdx.x;
  const float* inb  = in  + (size_t)b * CIN * HIN * HIN;
  float*       outb = out + (size_t)b * COUT * P;

  __shared__ float s_base[COUT];
  __shared__ float s_red[18];

  for (int c = tid; c < COUT; c += T) {
    float base = bias[c];
    if (HAS_LAB) {
      const float* lab = labels + b * 20;
#pragma unroll
      for (int lc = 0; lc < 20; ++lc) {
        const float4 wv = *reinterpret_cast<const float4*>(w + (size_t)(c * CTOT + CIN + lc) * 4);
        base += lab[lc] * (wv.x + wv.y + wv.z + wv.w);
      }
    }
    s_base[c] = base;
  }
  __syncthreads();

  const int wave = tid >> 5;
  const int lane = tid & 31;

  for (int c = 0; c < COUT; ++c) {
    const float* wc = w + (size_t)c * CTOT * 4;
    float vals[VPT];
    float s = 0.f, q = 0.f;
#pragma unroll
    for (int j = 0; j < VPT; ++j) {
      const int p = tid + j * T;
      const bool ok = (P % T == 0) || (p < P);
      float v = 0.f;
      if (ok) {
        v = s_base[c];
        const int oi = p / HOUT, oj = p % HOUT;
        const float* ip = inb + (size_t)(2 * oi) * HIN + 2 * oj;
#pragma unroll
        for (int ci = 0; ci < CIN; ++ci) {
          const float4 wv = *reinterpret_cast<const float4*>(wc + ci * 4);
          const v2f r0 = *reinterpret_cast<const v2f*>(ip + (size_t)ci * HIN * HIN);
          const v2f r1 = *reinterpret_cast<const v2f*>(ip + (size_t)ci * HIN * HIN + HIN);
          v += wv.x * r0.x + wv.y * r0.y + wv.z * r1.x + wv.w * r1.y;
        }
        s += v; q += v * v;
      }
      vals[j] = v;
    }
#pragma unroll
    for (int off = 16; off > 0; off >>= 1) {
      s += __shfl_down(s, off, 32);
      q += __shfl_down(q, off, 32);
    }
    if (lane == 0) { s_red[wave] = s; s_red[8 + wave] = q; }
    __syncthreads();
    if (tid == 0) {
      float S = 0.f, Q = 0.f;
#pragma unroll
      for (int i = 0; i < 8; ++i) { S += s_red[i]; Q += s_red[8 + i]; }
      const float mean = S / (float)P;
      const float var  = Q / (float)P - mean * mean;
      s_red[16] = mean;
      s_red[17] = rsqrtf(var + IN_EPS);
    }
    __syncthreads();
    const float mean = s_red[16];
    const float inv  = s_red[17];
    __syncthreads();
#pragma unroll
    for (int j = 0; j < VPT; ++j) {
      const int p = tid + j * T;
      if ((P % T == 0) || (p < P))
        outb[(size_t)c * P + p] = lrelu((vals[j] - mean) * inv);
    }
  }
}

// -------- Tensor Data Mover: 2D tile (rows x KB f32) global -> LDS ----------
// LDS rows padded to RS=68 dwords via D# pad (interval 64 dw -> code 5,
// amount 4 dw -> code 3) so b64 WMMA-operand reads are bank-conflict-free.
#define FC_KB 64
#define FC_RS 68

#if __has_builtin(__builtin_amdgcn_tensor_load_to_lds)
#define HAVE_TDM 1
__device__ __forceinline__ void tdm_load_2d(unsigned lds_off, const float* gptr,
                                            unsigned tile_rows)
{
  const unsigned long long ga = (unsigned long long)(size_t)gptr;
  v4u g0;
  g0[0] = 1u;                                             // count=1, user desc
  g0[1] = lds_off;                                        // lds_addr (bytes)
  g0[2] = (unsigned)(ga & 0xffffffffu);                   // global_addr lo
  g0[3] = (unsigned)((ga >> 32) & 0x01ffffffu) | (2u << 30); // addr hi | type=2
  v8i g1;
  // data_size=4B (2), pad_enable, pad_interval=64dw (5), pad_amount=4dw (3)
  g1[0] = (int)((2u << 16) | (1u << 20) | (5u << 22) | (3u << 25));
  g1[1] = (int)((unsigned)(FC_KB & 0xffff) << 16);        // tensor_dim0 lo16
  g1[2] = (int)((tile_rows & 0xffffu) << 16);             // dim0 hi=0 | dim1 lo
  g1[3] = (int)((unsigned)FC_KB << 16);                   // dim1 hi=0 | tile_dim0
  g1[4] = (int)tile_rows;                                 // tile_dim1, tile_dim2=0
  g1[5] = 8192;                                           // tensor_dim0_stride lo
  g1[6] = 0;                                              // stride hi | dim1_stride lo
  g1[7] = 0;
  const v4i g2 = {0, 0, 0, 0};
  const v4i g3 = {0, 0, 0, 0};
#if defined(__clang_major__) && (__clang_major__ >= 23)
  const v8i g4 = {0, 0, 0, 0, 0, 0, 0, 0};
  __builtin_amdgcn_tensor_load_to_lds(g0, g1, g2, g3, g4, 0);
#else
  __builtin_amdgcn_tensor_load_to_lds(g0, g1, g2, g3, 0);
#endif
}
#else
#define HAVE_TDM 0
#endif

// -------- FC1: [256,8192] x [8192,1024]^T via V_WMMA_F32_16X16X4_F32 --------
// Block = 8 waves = 128 M-rows x 16 N-cols macro tile; K staged in KB=64
// chunks through LDS (TDM, double buffered). All 8 waves share the W tile.
__global__ __launch_bounds__(256) void fc1_wmma_kernel(
    const float* __restrict__ A,     // [256, 8192]
    const float* __restrict__ W,     // [1024, 8192]
    const float* __restrict__ bias,  // [1024]
    float* __restrict__ O)           // [256, 1024]
{
  __shared__ float sA[2][128 * FC_RS];
  __shared__ float sW[2][16 * FC_RS];

  const int tid  = threadIdx.x;
  const int wave = tid >> 5;
  const int lane = tid & 31;
  const int half = lane >> 4;
  const int l    = lane & 15;
  const int nt   = blockIdx.x & 63;
  const int mb   = blockIdx.x >> 6;
  const int n0   = nt * 16;
  const int m0   = mb * 128;

  constexpr int NCHUNK = 8192 / FC_KB;

  v8f acc = {};
  int buf = 0;

#if HAVE_TDM
  if (wave == 0) {
    tdm_load_2d((unsigned)(size_t)&sA[0][0], A + (size_t)m0 * 8192, 128);
    tdm_load_2d((unsigned)(size_t)&sW[0][0], W + (size_t)n0 * 8192, 16);
  }
#endif

  for (int chunk = 0; chunk < NCHUNK; ++chunk) {
    const int knext = (chunk + 1) * FC_KB;
#if HAVE_TDM
    if (wave == 0) {
      if (chunk + 1 < NCHUNK) {
        tdm_load_2d((unsigned)(size_t)&sA[buf ^ 1][0], A + (size_t)m0 * 8192 + knext, 128);
        tdm_load_2d((unsigned)(size_t)&sW[buf ^ 1][0], W + (size_t)n0 * 8192 + knext, 16);
        __builtin_amdgcn_s_wait_tensorcnt(2);  // current chunk (older, in-order) done
      } else {
        __builtin_amdgcn_s_wait_tensorcnt(0);
      }
    }
#else
    {
      // Fallback staging: cooperative float4 copies into the padded layout.
      const int k0 = chunk * FC_KB;
      for (int idx = tid; idx < 128 * 16; idx += 256) {
        const int r = idx >> 4, f = idx & 15;
        *reinterpret_cast<float4*>(&sA[buf][r * FC_RS + f * 4]) =
            *reinterpret_cast<const float4*>(A + (size_t)(m0 + r) * 8192 + k0 + f * 4);
      }
      for (int idx = tid; idx < 16 * 16; idx += 256) {
        const int r = idx >> 4, f = idx & 15;
        *reinterpret_cast<float4*>(&sW[buf][r * FC_RS + f * 4]) =
            *reinterpret_cast<const float4*>(W + (size_t)(n0 + r) * 8192 + k0 + f * 4);
      }
    }
#endif
    __syncthreads();   // LDS tiles for `buf` visible to all waves

    const float* __restrict__ arow = &sA[buf][(wave * 16 + l) * FC_RS + 2 * half];
    const float* __restrict__ wrow = &sW[buf][l * FC_RS + 2 * half];
#pragma unroll
    for (int k = 0; k < FC_KB; k += 4) {
      const v2f av = *reinterpret_cast<const v2f*>(arow + k);
      const v2f bv = *reinterpret_cast<const v2f*>(wrow + k);
      acc = __builtin_amdgcn_wmma_f32_16x16x4_f32(false, av, false, bv,
                                                  (short)0, acc, false, false);
    }
    __syncthreads();   // all reads of `buf` done before it is refilled
    buf ^= 1;
  }

#pragma unroll
  for (int r = 0; r < 8; ++r) {
    const int m = m0 + wave * 16 + r + 8 * half;
    const int n = n0 + l;
    O[(size_t)m * 1024 + n] = lrelu(acc[r] + bias[n]);
  }
}

// -------- FC2: [256,1024] x [1024,1] ----------------------------------------
__global__ __launch_bounds__(256) void fc2_kernel(
    const float* __restrict__ H, const float* __restrict__ W2,
    const float* __restrict__ b2, float* __restrict__ out)
{
  __shared__ float red[256];
  const int bi = blockIdx.x, tid = threadIdx.x;
  float s = 0.f;
  for (int j = tid; j < 1024; j += 256) s += H[(size_t)bi * 1024 + j] * W2[j];
  red[tid] = s;
  __syncthreads();
  for (int off = 128; off > 0; off >>= 1) {
    if (tid < off) red[tid] += red[tid + off];
    __syncthreads();
  }
  if (tid == 0) out[bi] = red[0] + b2[0];
}

extern "C" void kernel_launch(void* const* d_in, const int* in_sizes, int n_in,
                              void* d_out, int out_size, void* d_ws, size_t ws_size,
                              hipStream_t stream) {
  (void)in_sizes; (void)n_in; (void)out_size; (void)ws_size;
  const float* x    = (const float*)d_in[0];
  const float* lab  = (const float*)d_in[1];
  const float* w1   = (const float*)d_in[2];
  const float* b1   = (const float*)d_in[3];
  const float* w2   = (const float*)d_in[4];
  const float* b2   = (const float*)d_in[5];
  const float* w3   = (const float*)d_in[6];
  const float* b3   = (const float*)d_in[7];
  const float* w4   = (const float*)d_in[8];
  const float* b4   = (const float*)d_in[9];
  const float* fcw1 = (const float*)d_in[10];
  const float* fcb1 = (const float*)d_in[11];
  const float* fcw2 = (const float*)d_in[12];
  const float* fcb2 = (const float*)d_in[13];

  float* ws = (float*)d_ws;
  float* h1 = ws;               // [256,16,64,64]  = 16,777,216 f
  float* h2 = ws + 16777216;    // [256,32,32,32]  =  8,388,608 f
  float* h3 = ws;               // [256,64,16,16]  =  4,194,304 f (reuse h1)
  float* h4 = ws + 4194304;     // [256,128,8,8]   =  2,097,152 f (after h3)
  float* h5 = ws + 16777216;    // [256,1024]      =    262,144 f (reuse h2)

  conv_in_lrelu_kernel< 3,  3,  16, 64, false><<<256, 256, 0, stream>>>(x,  w1, b1, nullptr, h1);
  conv_in_lrelu_kernel<16, 36,  32, 32, true ><<<256, 256, 0, stream>>>(h1, w2, b2, lab,     h2);
  conv_in_lrelu_kernel<32, 32,  64, 16, false><<<256, 256, 0, stream>>>(h2, w3, b3, nullptr, h3);
  conv_in_lrelu_kernel<64, 64, 128,  8, false><<<256, 256, 0, stream>>>(h3, w4, b4, nullptr, h4);
  fc1_wmma_kernel<<<128, 256, 0, stream>>>(h4, fcw1, fcb1, h5);
  fc2_kernel<<<256, 256, 0, stream>>>(h5, fcw2, fcb2, (float*)d_out);
}